// NeuralNetwork_28303834481424
// MI455X (gfx1250) — compile-verified
//
#include <hip/hip_runtime.h>
#include <math.h>

#define DLEN   17895
#define IVLEN  17932   // DLEN + 38 - 1
#define XCOLS  17935   // DLEN + 9 + 8 + 23
#define CHUNK  256
#define NCHUNK 71      // ceil(17932/256)
#define WSTR   298     // desc window stride: >=296, WSTR/2 odd -> conflict-free b64 reads
#define W0STR  268     // w0 stride: >=256, mult of 4 (b128 stores), (W0STR/2)%32 coprime-ish

typedef __attribute__((ext_vector_type(2))) float v2f;
typedef __attribute__((ext_vector_type(4))) float v4f;
typedef __attribute__((ext_vector_type(8))) float v8f;

__global__ __launch_bounds__(128)
void fused_conv_mlp(const float* __restrict__ x,
                    const float* __restrict__ W0,
                    const float* __restrict__ b0,
                    const float* __restrict__ W1,
                    const float* __restrict__ b1,
                    const float* __restrict__ W2,
                    const float* __restrict__ b2,
                    float* __restrict__ out)
{
  __shared__ float s_t2[16][40];
  __shared__ float s_desc[16][WSTR];
  __shared__ float s_w0[16][W0STR];
  __shared__ float s_acc[4][16][16];

  const int tid  = threadIdx.x;
  const int lane = tid & 31;
  const int wave = tid >> 5;
  const int r    = lane & 15;   // A: batch row in tile; B/C: feature column
  const int h    = lane >> 4;   // half-wave -> K offset 2h
  const int row0 = blockIdx.x << 4;

  // ---- Phase 0: per-row t2 = conv(glo, conv(p10, p11)), 38 taps ----
  if (tid < 16) {
    const float* xr = x + (size_t)(row0 + tid) * XCOLS;
    float g[9], pa[8], pb[23], t1[30];
    #pragma unroll
    for (int i = 0; i < 9;  ++i) g[i]  = xr[DLEN + i];
    #pragma unroll
    for (int i = 0; i < 8;  ++i) pa[i] = xr[DLEN + 9 + i];
    #pragma unroll
    for (int i = 0; i < 23; ++i) pb[i] = xr[DLEN + 17 + i];
    #pragma unroll
    for (int i = 0; i < 30; ++i) t1[i] = 0.f;
    for (int a = 0; a < 8; ++a)
      for (int b = 0; b < 23; ++b) t1[a + b] = fmaf(pa[a], pb[b], t1[a + b]);
    for (int i = 0; i < 38; ++i) {
      float s = 0.f;
      for (int a = 0; a < 9; ++a) {
        int j = i - a;
        if (j >= 0 && j < 30) s = fmaf(g[a], t1[j], s);
      }
      s_t2[tid][i] = s;
    }
  }
  __syncthreads();

  // ---- Per-lane reversed taps, pre-shifted into two aligned-pair arrays ----
  // u[j] = t2[37-j];  iv[n] = sum_j w[j]*u[j],  iv[n+1] = sum_j w[j]*u[j-1]
  float u[40];
  #pragma unroll
  for (int j = 0; j < 38; ++j) u[j] = s_t2[r][37 - j];
  u[38] = 0.f; u[39] = 0.f;
  v2f U0[20], U1[20];
  #pragma unroll
  for (int k = 0; k < 20; ++k) {
    U0[k].x = u[2 * k];                  U0[k].y = u[2 * k + 1];
    U1[k].x = (k == 0) ? 0.f : u[2 * k - 1];
    U1[k].y = u[2 * k];
  }

  v8f acc = {0.f, 0.f, 0.f, 0.f, 0.f, 0.f, 0.f, 0.f};

  // Cooperative loader mapping: 8 threads per row.
  const int lr = tid >> 3;
  const int lc = tid & 7;
  const float* dsrc = x  + (size_t)(row0 + lr) * XCOLS;  // desc of row lr
  const float* wsrc = W0 + (size_t)lr * IVLEN;           // W0 feature row lr

  for (int c = 0; c < NCHUNK; ++c) {
    const int nbase = c * CHUNK;

    // --- Stage desc window [nbase-37, nbase+258] (zero-padded full-conv edges) ---
    if (c == 0 || c >= 69) {
      for (int kk = lc; kk < 296; kk += 8) {
        int g2 = nbase - 37 + kk;
        s_desc[lr][kk] = (g2 >= 0 && g2 < DLEN) ? dsrc[g2] : 0.f;
      }
    } else {
      const float* p = dsrc + (nbase - 37 + lc);
      #pragma unroll
      for (int i = 0; i < 37; ++i) s_desc[lr][lc + 8 * i] = p[8 * i];
    }
    // --- Stage W0 column block [nbase, nbase+255] ---
    if (c == NCHUNK - 1) {
      for (int kk = lc; kk < CHUNK; kk += 8) {
        int g2 = nbase + kk;
        s_w0[lr][kk] = (g2 < IVLEN) ? wsrc[g2] : 0.f;
      }
    } else {
      const v4f* p4 = (const v4f*)(wsrc + nbase);       // 16B aligned
      v4f*       s4 = (v4f*)&s_w0[lr][0];               // row stride 268*4 = 16B mult
      #pragma unroll
      for (int i = 0; i < 8; ++i) s4[lc + 8 * i] = p4[lc + 8 * i];
    }
    __syncthreads();

    // --- Compute: rolling register window, 16 WMMA steps, all pairs aligned ---
    const int wbase = wave << 6;                         // this wave's 64 n's
    const v2f*   wp = (const v2f*)&s_desc[r][wbase + (h << 1)];
    const float* bp = &s_w0[r][wbase + (h << 1)];

    v2f w2[52];
    #pragma unroll
    for (int k = 0; k < 20; ++k) w2[k] = wp[k];

    #pragma unroll
    for (int s = 0; s < 16; ++s) {
      if (s > 0) {                                       // slide: 2 new b64 loads
        w2[2 * s + 18] = wp[2 * s + 18];
        w2[2 * s + 19] = wp[2 * s + 19];
      }
      v2f A0 = {0.f, 0.f}, A1 = {0.f, 0.f};
      #pragma unroll
      for (int k = 0; k < 20; ++k) {                     // 40 aligned v_pk_fma_f32
        A0 += w2[2 * s + k] * U0[k];
        A1 += w2[2 * s + k] * U1[k];
      }
      v2f A;  A.x = A0.x + A0.y;  A.y = A1.x + A1.y;     // iv pair for this lane
      v2f Bv = *(const v2f*)(bp + 4 * s);                // W0 pair, same n's
      acc = __builtin_amdgcn_wmma_f32_16x16x4_f32(false, A, false, Bv,
                                                  (short)0, acc, false, false);
    }
    __syncthreads();
  }

  // Spill per-wave C (ISA layout: VGPR v -> M = v + 8h, N = r)
  #pragma unroll
  for (int v = 0; v < 8; ++v)
    s_acc[wave][v + (h << 3)][r] = acc[v];
  __syncthreads();

  // Reduce 4 waves + MLP head, one thread per batch row
  if (tid < 16) {
    float hv[16];
    #pragma unroll
    for (int j = 0; j < 16; ++j) {
      float s = s_acc[0][tid][j] + s_acc[1][tid][j] +
                s_acc[2][tid][j] + s_acc[3][tid][j] + b0[j];
      hv[j] = (s > 0.f) ? s : (expf(s) - 1.f);   // ELU
    }
    float o = b2[0];
    #pragma unroll
    for (int q = 0; q < 4; ++q) {
      float s = b1[q];
      #pragma unroll
      for (int j = 0; j < 16; ++j) s = fmaf(hv[j], W1[q * 16 + j], s);
      s = (s > 0.f) ? s : (expf(s) - 1.f);       // ELU
      o = fmaf(s, W2[q], o);
    }
    out[row0 + tid] = o;
  }
}

extern "C" void kernel_launch(void* const* d_in, const int* in_sizes, int n_in,
                              void* d_out, int out_size, void* d_ws, size_t ws_size,
                              hipStream_t stream) {
  const float* x  = (const float*)d_in[0];
  const float* W0 = (const float*)d_in[1];
  const float* b0 = (const float*)d_in[2];
  const float* W1 = (const float*)d_in[3];
  const float* b1 = (const float*)d_in[4];
  const float* W2 = (const float*)d_in[5];
  const float* b2 = (const float*)d_in[6];
  float* out = (float*)d_out;

  const int B = in_sizes[0] / XCOLS;    // 4096
  dim3 grid(B / 16), block(128);
  hipLaunchKernelGGL(fused_conv_mlp, grid, block, 0, stream,
                     x, W0, b0, W1, b1, W2, b2, out);
}